// ChildSumTreeLSTM_40243843564225
// MI455X (gfx1250) — compile-verified
//
#include <hip/hip_runtime.h>
#include <hip/hip_bf16.h>
#include <math.h>

// ---------------------------------------------------------------------------
// ChildSumTreeLSTM on gfx1250 (MI455X)
// Tree is complete & level-ordered -> 10 level-parallel steps.
// GEMMs: V_WMMA_F32_16X16X4_F32, 8 waves/block share an LDS-staged B panel.
// Inner loop: 1x global_load_b64 (A) + 1x ds_load_b64 (B) + 1x v_wmma.
// ---------------------------------------------------------------------------

#define NNODES 1023
#define HDIM   300
#define EDIM   300
#define NCOLS  900          // 3*H concatenated [W_i* | W_f* | W_u*] outputs
#define GSTRIDE 912         // padded row stride for G (57*16)
#define LBL    5
#define KLEN   300
#define PANEL  (150 * 32)   // 300x16 B panel, interleaved (k-pair, n) layout

typedef float v2f __attribute__((ext_vector_type(2)));
typedef float v8f __attribute__((ext_vector_type(8)));

__device__ __forceinline__ v8f wmma_f32_k4(v2f a, v2f b, v8f c) {
    // 8 args: (neg_a, A, neg_b, B, c_mod, C, reuse_a, reuse_b)
    return __builtin_amdgcn_wmma_f32_16x16x4_f32(
        false, a, false, b, (short)0, c, false, false);
}

__device__ __forceinline__ float sigmoidf_(float x) {
    return 1.0f / (1.0f + expf(-x));
}

// Stage B panel for output-column tile tn into LDS.
// Layout: lb[(k>>1)*32 + 2*n + (k&1)]  ->  lane fragment (B[kA][n],B[kA+1][n])
// is one aligned 8-byte LDS load at float index 64*i0 + 32*half + 2*n.
__device__ __forceinline__ void stage_B(float* lb, int tn,
                                        const float* __restrict__ W0,
                                        const float* __restrict__ W1,
                                        const float* __restrict__ W2) {
    for (int e = threadIdx.x; e < KLEN * 16; e += 256) {
        const int k = e >> 4;
        const int n = e & 15;
        const int j = tn * 16 + n;
        const float* wsel;
        int jj;
        if (j < 300)      { wsel = W0; jj = j; }
        else if (j < 600) { wsel = W1; jj = j - 300; }
        else if (j < 900) { wsel = W2; jj = j - 600; }
        else              { wsel = W0; jj = 0; }   // safe dummy; col never stored
        lb[((k >> 1) << 5) + (n << 1) + (k & 1)] = wsel[k * 300 + jj];
    }
}

// ---------------------------------------------------------------------------
// Phase 1: XP[t][0:900] = emb[word_ids[t]] @ [W_ix|W_fx|W_ux] + [b_ix|b_fx|b_ux]
// block = 256 (8 waves, one M-tile each, shared N-tile). grid = (8, 57).
// ---------------------------------------------------------------------------
__global__ void xproj_gemm(const int* __restrict__ word_ids,
                           const float* __restrict__ emb,
                           const float* __restrict__ Wix, const float* __restrict__ bix,
                           const float* __restrict__ Wfx, const float* __restrict__ bfx,
                           const float* __restrict__ Wux, const float* __restrict__ bux,
                           float* __restrict__ XP) {
    __shared__ float lb[PANEL];
    const int tn = blockIdx.y;
    stage_B(lb, tn, Wix, Wfx, Wux);
    __syncthreads();

    const int w    = threadIdx.x >> 5;
    const int l    = threadIdx.x & 31;
    const int half = l >> 4;
    const int lane = l & 15;
    const int tm   = blockIdx.x * 8 + w;

    // A fragment: row m of gathered embedding (dummy row 0 if out of range —
    // those D rows are never stored).
    const int m = tm * 16 + lane;
    const float* arow = emb + (size_t)word_ids[(m < NNODES) ? m : 0] * (size_t)EDIM;

    const float* lbb = lb + 32 * half + 2 * lane;
    v8f acc = {};
    #pragma unroll 5
    for (int i0 = 0; i0 < KLEN / 4; ++i0) {
        const v2f a = *(const v2f*)(arow + 4 * i0 + 2 * half);
        const v2f b = *(const v2f*)(lbb + 64 * i0);
        acc = wmma_f32_k4(a, b, acc);
    }

    const int j = tn * 16 + lane;
    if (j < NCOLS) {
        const float* bsel;
        int jj;
        if (j < 300)      { bsel = bix; jj = j; }
        else if (j < 600) { bsel = bfx; jj = j - 300; }
        else              { bsel = bux; jj = j - 600; }
        const float bias = bsel[jj];
        #pragma unroll
        for (int v = 0; v < 8; ++v) {
            const int mm = tm * 16 + v + 8 * half;
            if (mm < NNODES) XP[(size_t)mm * NCOLS + j] = acc[v] + bias;
        }
    }
}

// ---------------------------------------------------------------------------
// Per level: G[r][0:900] = mask_r * (h_all[child(r)] @ [W_ih|W_fh|W_uh])
// Mask applied at store (exact for 0/1 masks) so the K-loop stays pure.
// ---------------------------------------------------------------------------
__global__ void child_gemm(const int* __restrict__ children_idx,
                           const int* __restrict__ children_mask,
                           const float* __restrict__ h_all,
                           const float* __restrict__ Wih,
                           const float* __restrict__ Wfh,
                           const float* __restrict__ Wuh,
                           float* __restrict__ G,
                           int start, int size) {
    __shared__ float lb[PANEL];
    const int tn = blockIdx.y;
    stage_B(lb, tn, Wih, Wfh, Wuh);
    __syncthreads();

    const int w    = threadIdx.x >> 5;
    const int l    = threadIdx.x & 31;
    const int half = l >> 4;
    const int lane = l & 15;
    const int tm   = blockIdx.x * 8 + w;
    const int R    = 2 * size;

    const int r  = tm * 16 + lane;
    const int rc = (r < R) ? r : 0;                 // dummy row if out of range
    const int node = start + (rc >> 1);
    const int ci   = children_idx[2 * node + (rc & 1)];
    const float* arow = h_all + (size_t)ci * HDIM;

    const float* lbb = lb + 32 * half + 2 * lane;
    v8f acc = {};
    #pragma unroll 5
    for (int i0 = 0; i0 < KLEN / 4; ++i0) {
        const v2f a = *(const v2f*)(arow + 4 * i0 + 2 * half);
        const v2f b = *(const v2f*)(lbb + 64 * i0);
        acc = wmma_f32_k4(a, b, acc);
    }

    const int j = tn * 16 + lane;
    if (j < NCOLS) {
        #pragma unroll
        for (int v = 0; v < 8; ++v) {
            const int rr = tm * 16 + v + 8 * half;
            if (rr < R) {
                const int rn = start + (rr >> 1);
                const float mrr = (float)children_mask[2 * rn + (rr & 1)];
                G[(size_t)rr * GSTRIDE + j] = acc[v] * mrr;
            }
        }
    }
}

// ---------------------------------------------------------------------------
// Cell: gates + state update for one level (size*300 elements).
//   i = sig(ixx + hsum@W_ih + b_ih)
//   o = sig(fxx + hsum@W_fh + b_fh)        (reference quirk: o uses fxx/W_fh)
//   u = sig(uxx + hsum@W_uh + b_uh)        (sigmoid, not tanh)
//   f_k = sig(ch_h_k@W_fh + b_fh + fxx)
//   c = i*u + sum_k f_k * (c_child_k * mask_k)
//   h = o * tanh(c)
// ---------------------------------------------------------------------------
__global__ void cell_kernel(const float* __restrict__ XP,
                            const float* __restrict__ G,
                            const int* __restrict__ children_idx,
                            const int* __restrict__ children_mask,
                            const float* __restrict__ bih,
                            const float* __restrict__ bfh,
                            const float* __restrict__ buh,
                            float* __restrict__ h_all,
                            float* __restrict__ c_all,
                            int start, int size, int interior) {
    const int idx = blockIdx.x * blockDim.x + threadIdx.x;
    if (idx >= size * HDIM) return;
    const int r = idx / HDIM;
    const int j = idx - r * HDIM;
    const int t = start + r;

    const float xpi = XP[(size_t)t * NCOLS + j];
    const float xpf = XP[(size_t)t * NCOLS + 300 + j];
    const float xpu = XP[(size_t)t * NCOLS + 600 + j];
    const float bi = bih[j], bf = bfh[j], bu = buh[j];

    float i_, o_, u_, c_;
    if (interior) {
        const float* g0 = G + (size_t)(2 * r)     * GSTRIDE;
        const float* g1 = G + (size_t)(2 * r + 1) * GSTRIDE;
        const float g0i = g0[j],       g1i = g1[j];
        const float g0f = g0[300 + j], g1f = g1[300 + j];
        const float g0u = g0[600 + j], g1u = g1[600 + j];

        i_ = sigmoidf_(xpi + g0i + g1i + bi);
        o_ = sigmoidf_(xpf + g0f + g1f + bf);
        u_ = sigmoidf_(xpu + g0u + g1u + bu);
        const float f0 = sigmoidf_(g0f + bf + xpf);
        const float f1 = sigmoidf_(g1f + bf + xpf);

        const int c0 = children_idx[2 * t];
        const int c1 = children_idx[2 * t + 1];
        const float m0 = (float)children_mask[2 * t];
        const float m1 = (float)children_mask[2 * t + 1];
        const float cc0 = c_all[(size_t)c0 * HDIM + j] * m0;
        const float cc1 = c_all[(size_t)c1 * HDIM + j] * m1;
        c_ = i_ * u_ + f0 * cc0 + f1 * cc1;
    } else {
        i_ = sigmoidf_(xpi + bi);
        o_ = sigmoidf_(xpf + bf);
        u_ = sigmoidf_(xpu + bu);
        c_ = i_ * u_;
    }
    const float h_ = o_ * tanhf(c_);
    c_all[(size_t)t * HDIM + j] = c_;
    h_all[(size_t)t * HDIM + j] = h_;
}

// ---------------------------------------------------------------------------
// Output: logits = h_all @ W_out + b_out ; logp = log_softmax ; loss.
// Single 1024-thread block; deterministic LDS tree reduction (no atomics).
// ---------------------------------------------------------------------------
__global__ void output_kernel(const float* __restrict__ h_all,
                              const float* __restrict__ W_out,
                              const float* __restrict__ b_out,
                              const int* __restrict__ labels,
                              float* __restrict__ out) {
    __shared__ float red[1024];
    const int t = threadIdx.x;
    float lossc = 0.0f;
    if (t < NNODES) {
        float lg[LBL];
        #pragma unroll
        for (int l2 = 0; l2 < LBL; ++l2) lg[l2] = b_out[l2];
        const float* hrow = h_all + (size_t)t * HDIM;
        for (int k = 0; k < HDIM; ++k) {
            const float hv = hrow[k];
            #pragma unroll
            for (int l2 = 0; l2 < LBL; ++l2) lg[l2] += hv * W_out[k * LBL + l2];
        }
        float mx = lg[0];
        #pragma unroll
        for (int l2 = 1; l2 < LBL; ++l2) mx = fmaxf(mx, lg[l2]);
        float s = 0.0f;
        #pragma unroll
        for (int l2 = 0; l2 < LBL; ++l2) s += expf(lg[l2] - mx);
        const float lz = mx + logf(s);
        #pragma unroll
        for (int l2 = 0; l2 < LBL; ++l2) out[t * LBL + l2] = lg[l2] - lz;
        lossc = -(lg[labels[t]] - lz);
    }
    red[threadIdx.x] = lossc;
    __syncthreads();
    for (int off = 512; off > 0; off >>= 1) {
        if (threadIdx.x < off) red[threadIdx.x] += red[threadIdx.x + off];
        __syncthreads();
    }
    if (threadIdx.x == 0) out[NNODES * LBL] = red[0];
}

// ---------------------------------------------------------------------------
extern "C" void kernel_launch(void* const* d_in, const int* in_sizes, int n_in,
                              void* d_out, int out_size, void* d_ws, size_t ws_size,
                              hipStream_t stream) {
    const int*   word_ids      = (const int*)d_in[0];
    const int*   labels        = (const int*)d_in[1];
    const int*   children_idx  = (const int*)d_in[2];
    const int*   children_mask = (const int*)d_in[3];
    const float* emb   = (const float*)d_in[4];
    const float* W_ix  = (const float*)d_in[5];  const float* b_ix = (const float*)d_in[6];
    const float* W_ih  = (const float*)d_in[7];  const float* b_ih = (const float*)d_in[8];
    const float* W_fx  = (const float*)d_in[9];  const float* b_fx = (const float*)d_in[10];
    const float* W_fh  = (const float*)d_in[11]; const float* b_fh = (const float*)d_in[12];
    const float* W_ux  = (const float*)d_in[13]; const float* b_ux = (const float*)d_in[14];
    const float* W_uh  = (const float*)d_in[15]; const float* b_uh = (const float*)d_in[16];
    const float* W_out = (const float*)d_in[17]; const float* b_out = (const float*)d_in[18];

    float* out = (float*)d_out;
    float* ws  = (float*)d_ws;

    // workspace layout (floats)
    float* XP    = ws;                      // 1023 * 900 = 920700
    float* G     = ws + 920700;             //  512 * 912 = 466944
    float* h_all = ws + 920700 + 466944;    // 1023 * 300 = 306900
    float* c_all = h_all + 306900;          // 1023 * 300 = 306900

    // Phase 1: embedding gather + x-projections (WMMA fp32)
    xproj_gemm<<<dim3(8, 57), 256, 0, stream>>>(
        word_ids, emb, W_ix, b_ix, W_fx, b_fx, W_ux, b_ux, XP);

    // Levels: (start,size) = (0,512),(512,256),...,(1022,1)
    int start = 0, size = 512;
    for (int lvl = 0; lvl < 10; ++lvl) {
        if (lvl > 0) {
            const int R = 2 * size;
            const int mtiles = (R + 15) / 16;
            child_gemm<<<dim3((mtiles + 7) / 8, 57), 256, 0, stream>>>(
                children_idx, children_mask, h_all, W_ih, W_fh, W_uh, G, start, size);
        }
        const int tot = size * HDIM;
        cell_kernel<<<(tot + 255) / 256, 256, 0, stream>>>(
            XP, G, children_idx, children_mask, b_ih, b_fh, b_uh,
            h_all, c_all, start, size, lvl > 0 ? 1 : 0);
        start += size;
        size >>= 1;
    }

    // Output projection + log_softmax + loss
    output_kernel<<<1, 1024, 0, stream>>>(h_all, W_out, b_out, labels, out);
}